// LMMSE_22419729285621
// MI455X (gfx1250) — compile-verified
//
#include <hip/hip_runtime.h>

typedef __attribute__((ext_vector_type(2))) float v2f;
typedef __attribute__((ext_vector_type(4))) float v4f;
typedef __attribute__((ext_vector_type(8))) float v8f;

namespace {
constexpr int  kB   = 16;
constexpr int  kTX  = 4;
constexpr int  kS   = 3276;
constexpr int  kT   = 14;
constexpr int  kST  = kS * kT;                 // 45864
constexpr long kNRE = (long)kB * kST;          // 733824
constexpr long kNALL= (long)kB * kTX * kST;    // 2935296
constexpr float kEPS = 1e-5f;
}

#if defined(__AMDGCN__) && __has_builtin(__builtin_amdgcn_wmma_f32_16x16x4_f32)
#define LMMSE_USE_WMMA 1
#else
#define LMMSE_USE_WMMA 0
#endif

__global__ __launch_bounds__(256)
void lmmse_wmma_kernel(const float* __restrict__ yre, const float* __restrict__ yim,
                       const float* __restrict__ hh,  const float* __restrict__ srl,
                       const float* __restrict__ act,
                       const float* __restrict__ gamma_p, const float* __restrict__ theta_p,
                       const float* __restrict__ zeta_p,
                       float* __restrict__ out)
{
#if LMMSE_USE_WMMA
    __shared__ float hS[8][32][32];   // per-wave: h staging rows, reused for Gram scatter
#endif
    const int tid  = threadIdx.x;
    const int wave = tid >> 5;
    const int lane = tid & 31;
    (void)wave; (void)lane;

    long e = (long)blockIdx.x * 256 + tid;
    const bool valid = (e < kNRE);
    if (!valid) e = kNRE - 1;          // clamp: keep EXEC all-ones for WMMA + uniform barriers
    const int b  = (int)(e / kST);
    const int st = (int)(e % kST);

    const float gmma = gamma_p[0];
    const float thta = theta_p[0];
    const float zeta = zeta_p[0];

    const long act_base = ((long)b * kTX) * kST + st;
    __builtin_prefetch(act + act_base, 0, 1);          // global_prefetch_b8

    // ---- stream in h (R x TX complex), y, s ----
    float hr_[4][4], hi_[4][4];        // [row r][col tx]
    const float* hbase = hh + (((long)b * kTX) * kST + st) * 8;
    #pragma unroll
    for (int tx = 0; tx < 4; ++tx) {
        const v4f* hp = (const v4f*)(hbase + (long)tx * kST * 8);
        v4f a = __builtin_nontemporal_load(hp);
        v4f c = __builtin_nontemporal_load(hp + 1);
        hr_[0][tx]=a.x; hr_[1][tx]=a.y; hr_[2][tx]=a.z; hr_[3][tx]=a.w;
        hi_[0][tx]=c.x; hi_[1][tx]=c.y; hi_[2][tx]=c.z; hi_[3][tx]=c.w;
#if LMMSE_USE_WMMA
        v4f* dst = (v4f*)&hS[wave][lane][tx * 8];
        dst[0] = a; dst[1] = c;
#endif
    }

    float yv[4], yw[4];
    {
        v4f a = __builtin_nontemporal_load((const v4f*)(yre + e * 4));
        v4f c = __builtin_nontemporal_load((const v4f*)(yim + e * 4));
        yv[0]=a.x; yv[1]=a.y; yv[2]=a.z; yv[3]=a.w;
        yw[0]=c.x; yw[1]=c.y; yw[2]=c.z; yw[3]=c.w;
    }

    float sv[4][4];
    {
        const v4f* sp = (const v4f*)(srl + e * 16);
        #pragma unroll
        for (int i = 0; i < 4; ++i) {
            v4f a = __builtin_nontemporal_load(sp + i);
            sv[i][0] = fmaxf(gmma * a.x, 0.f) + kEPS;
            sv[i][1] = fmaxf(gmma * a.y, 0.f) + kEPS;
            sv[i][2] = fmaxf(gmma * a.z, 0.f) + kEPS;
            sv[i][3] = fmaxf(gmma * a.w, 0.f) + kEPS;
        }
    }

    float gRe[4][4], gIm[4][4];
#if LMMSE_USE_WMMA
    __syncthreads();
    {
        // A 16x4 f32 fragment layout: lanes 0-15 M=lane (K=0 in v0, K=1 in v1),
        // lanes 16-31 M=lane-16 (K=2,3). B 4x16 mirrors it -> B frags == A frags here.
        const int half = lane >> 4;
        const int l16  = lane & 15;
        const int ai   = l16 >> 2;           // RE-in-group (row/col block)
        const int ar   = l16 & 3;            // row within block
        const int k0   = half * 2;
        const int k1   = k0 + 1;
        // D scatter decode: block (i,i) of RE i lives in lanes {4i+c} (i<2) / {16+4i+c} (i>=2)
        const bool useful = (half == (l16 >> 3));
        const int  si  = l16 >> 2;
        const int  sc  = l16 & 3;
        const int  hi4 = si & 1;

        #pragma unroll
        for (int q = 0; q < 8; ++q) {        // 8 groups of 4 REs per wave
            const float* hwp = &hS[wave][(q << 2) + ai][0];
            v2f Are, Aim;
            Are.x = hwp[(k0 << 3) + ar];
            Are.y = hwp[(k1 << 3) + ar];
            Aim.x = hwp[(k0 << 3) + 4 + ar];
            Aim.y = hwp[(k1 << 3) + 4 + ar];
            v2f Ain; Ain.x = -Aim.x; Ain.y = -Aim.y;
            v8f zero8 = {0.f,0.f,0.f,0.f,0.f,0.f,0.f,0.f};
            // Re(G) = Hr*Hr^T + Hi*Hi^T ; Im(G) = Hi*Hr^T - Hr*Hi^T  (exact fp32 WMMA)
            v8f t0  = __builtin_amdgcn_wmma_f32_16x16x4_f32(false, Aim, false, Aim, (short)0, zero8, false, false);
            v8f wRe = __builtin_amdgcn_wmma_f32_16x16x4_f32(false, Are, false, Are, (short)0, t0,    false, false);
            v8f t1  = __builtin_amdgcn_wmma_f32_16x16x4_f32(false, Are, false, Ain, (short)0, zero8, false, false);
            v8f wIm = __builtin_amdgcn_wmma_f32_16x16x4_f32(false, Aim, false, Are, (short)0, t1,    false, false);
            if (useful) {
                // write column sc of RE (q*4+si): Re at [sc*4..], Im at [16+sc*4..]
                v4f* gp = (v4f*)&hS[wave][(q << 2) + si][0];
                if (hi4 == 0) {
                    v4f r4; r4.x = wRe[0]; r4.y = wRe[1]; r4.z = wRe[2]; r4.w = wRe[3];
                    v4f m4; m4.x = wIm[0]; m4.y = wIm[1]; m4.z = wIm[2]; m4.w = wIm[3];
                    gp[sc]     = r4;
                    gp[4 + sc] = m4;
                } else {
                    v4f r4; r4.x = wRe[4]; r4.y = wRe[5]; r4.z = wRe[6]; r4.w = wRe[7];
                    v4f m4; m4.x = wIm[4]; m4.y = wIm[5]; m4.z = wIm[6]; m4.w = wIm[7];
                    gp[sc]     = r4;
                    gp[4 + sc] = m4;
                }
            }
        }
    }
    __syncthreads();
    {
        const v4f* gp4 = (const v4f*)&hS[wave][lane][0];
        #pragma unroll
        for (int c = 0; c < 4; ++c) {
            v4f a = gp4[c];
            gRe[0][c]=a.x; gRe[1][c]=a.y; gRe[2][c]=a.z; gRe[3][c]=a.w;
            v4f m = gp4[4 + c];
            gIm[0][c]=m.x; gIm[1][c]=m.y; gIm[2][c]=m.z; gIm[3][c]=m.w;
        }
    }
#else
    #pragma unroll
    for (int i = 0; i < 4; ++i)
        #pragma unroll
        for (int j = 0; j < 4; ++j) {
            float re = 0.f, im = 0.f;
            #pragma unroll
            for (int k = 0; k < 4; ++k) {
                re += hr_[i][k]*hr_[j][k] + hi_[i][k]*hi_[j][k];
                im += hi_[i][k]*hr_[j][k] - hr_[i][k]*hi_[j][k];
            }
            gRe[i][j] = re; gIm[i][j] = im;
        }
#endif

    // m = 0.5*(m + m^H) with m = g + s  (s real, relu(gamma*s)+eps already applied)
    float Gr[4][4], Gi[4][4];
    #pragma unroll
    for (int i = 0; i < 4; ++i)
        #pragma unroll
        for (int j = 0; j < 4; ++j) {
            Gr[i][j] = 0.5f * (gRe[i][j] + gRe[j][i] + sv[i][j] + sv[j][i]);
            Gi[i][j] = 0.5f * (gIm[i][j] - gIm[j][i]);
        }

    // ---- complex Cholesky G = L L^H (divergence-free, fully unrolled) ----
    float Lr[4][4], Li[4][4], Ldi[4];
    #pragma unroll
    for (int j = 0; j < 4; ++j) {
        float d = Gr[j][j];
        #pragma unroll
        for (int k = 0; k < j; ++k) d -= Lr[j][k]*Lr[j][k] + Li[j][k]*Li[j][k];
        float sdg = sqrtf(d);
        float inv = 1.0f / sdg;
        Lr[j][j] = sdg; Li[j][j] = 0.f; Ldi[j] = inv;
        #pragma unroll
        for (int i = j + 1; i < 4; ++i) {
            float ar = Gr[i][j], aii = Gi[i][j];
            #pragma unroll
            for (int k = 0; k < j; ++k) {
                ar  -= Lr[i][k]*Lr[j][k] + Li[i][k]*Li[j][k];
                aii -= Li[i][k]*Lr[j][k] - Lr[i][k]*Li[j][k];
            }
            Lr[i][j] = ar * inv; Li[i][j] = aii * inv;
        }
    }

    // ---- X = G^{-1} h  (forward + backward substitution); w = X^H ----
    float Xr[4][4], Xi[4][4];
    #pragma unroll
    for (int c = 0; c < 4; ++c) {
        #pragma unroll
        for (int i = 0; i < 4; ++i) {
            float ar = hr_[i][c], aii = hi_[i][c];
            #pragma unroll
            for (int k = 0; k < i; ++k) {
                ar  -= Lr[i][k]*Xr[k][c] - Li[i][k]*Xi[k][c];
                aii -= Lr[i][k]*Xi[k][c] + Li[i][k]*Xr[k][c];
            }
            Xr[i][c] = ar * Ldi[i]; Xi[i][c] = aii * Ldi[i];
        }
        #pragma unroll
        for (int i = 3; i >= 0; --i) {
            float ar = Xr[i][c], aii = Xi[i][c];
            #pragma unroll
            for (int k = i + 1; k < 4; ++k) {
                ar  -= Lr[k][i]*Xr[k][c] + Li[k][i]*Xi[k][c];
                aii -= Lr[k][i]*Xi[k][c] - Li[k][i]*Xr[k][c];
            }
            Xr[i][c] = ar * Ldi[i]; Xi[i][c] = aii * Ldi[i];
        }
    }

    // ---- gy = w y, d = diag(w h), outputs ----
    if (valid) {
        #pragma unroll
        for (int tx = 0; tx < 4; ++tx) {
            float gyr = 0.f, gyi = 0.f, ddr = 0.f, ddi = 0.f;
            #pragma unroll
            for (int r = 0; r < 4; ++r) {
                float xr = Xr[r][tx], xi = Xi[r][tx];   // w[tx][r] = conj(X[r][tx])
                gyr += xr*yv[r]       + xi*yw[r];
                gyi += xr*yw[r]       - xi*yv[r];
                ddr += xr*hr_[r][tx]  + xi*hi_[r][tx];
                ddi += xr*hi_[r][tx]  - xi*hr_[r][tx];
            }
            const long oidx = act_base + (long)tx * kST;
            float xre = 0.f, xim = 0.f;
            if ((ddr != 0.f) || (ddi != 0.f)) {
                float invden = 1.0f / (ddr*ddr + ddi*ddi);
                xre = (gyr*ddr + gyi*ddi) * invden;
                xim = (gyi*ddr - gyr*ddi) * invden;
            }
            float am = zeta * act[oidx];
            out[2*oidx]     = am * xre;
            out[2*oidx + 1] = am * xim;
            float ninv = (ddr != 0.f) ? (1.0f / ddr) : 0.f;
            out[2*kNALL + oidx] = fmaxf(thta * (ninv - 1.0f), 0.f) + kEPS;
        }
    }
}

extern "C" void kernel_launch(void* const* d_in, const int* in_sizes, int n_in,
                              void* d_out, int out_size, void* d_ws, size_t ws_size,
                              hipStream_t stream) {
    (void)in_sizes; (void)n_in; (void)out_size; (void)d_ws; (void)ws_size;
    const float* yre   = (const float*)d_in[0];
    const float* yim   = (const float*)d_in[1];
    const float* hh    = (const float*)d_in[2];
    const float* srl   = (const float*)d_in[3];
    const float* act   = (const float*)d_in[4];
    // d_in[5] = mcs_ue_mask: unused by the reference computation
    const float* gamma = (const float*)d_in[6];
    const float* theta = (const float*)d_in[7];
    const float* zeta  = (const float*)d_in[8];
    float* out = (float*)d_out;

    const int nblk = (int)((kNRE + 255) / 256);   // 2867 blocks of 8 wave32
    hipLaunchKernelGGL(lmmse_wmma_kernel, dim3(nblk), dim3(256), 0, stream,
                       yre, yim, hh, srl, act, gamma, theta, zeta, out);
}